// Yolov1_Loss_40450001994029
// MI455X (gfx1250) — compile-verified
//
#include <hip/hip_runtime.h>
#include <hip/hip_bf16.h>
#include <stdint.h>

// ---------------------------------------------------------------------------
// YOLOv1 loss on MI455X (gfx1250).
// Memory-bound streaming reduction (~226 MB in, 6 scalars out, ~10us at
// 23.3 TB/s). Data path: Tensor Data Mover (tensor_load_to_lds) double-
// buffered tiles into LDS, wave32 shfl reduction, deterministic 2-pass sum.
// ---------------------------------------------------------------------------

#define S_DIM            12
#define CELLS_PER_IMG    144          // S*S
#define FLOATS_PER_CELL  12           // both bb and gt have 12 f32 per cell
#define TILE_CELLS       256          // == blockDim.x, 1 cell per thread
#define TILE_FLOATS      (TILE_CELLS * FLOATS_PER_CELL)   // 3072
#define TILE_BYTES       (TILE_FLOATS * 4)                // 12288
#define TILES_PER_BLOCK  8
#define MAIN_BLOCKS      1152         // 1152*8*256 = 2,359,296 = BATCH*S*S
#define NACC             8            // 7 partial sums padded to 8

typedef __attribute__((ext_vector_type(4))) unsigned int u32x4;
typedef __attribute__((ext_vector_type(8))) unsigned int u32x8;

// Issue a 1-D TDM copy of TILE_FLOATS f32 elements: global -> LDS.
// D# group0: count=1 | lds_addr | global_addr[56:0] | type=2.
// Shared group1 (g1) carries data_size=4B, tensor_dim0=tile_dim0=3072.
__device__ __forceinline__ void tdm_load_1d(uint64_t gaddr, uint32_t lds_byte,
                                            u32x8 g1) {
  u32x4 g0;
  g0.x = 1u;                                               // count=1, user D#
  g0.y = (uint32_t)__builtin_amdgcn_readfirstlane((int)lds_byte);
  g0.z = (uint32_t)__builtin_amdgcn_readfirstlane((int)(uint32_t)gaddr);
  g0.w = (uint32_t)__builtin_amdgcn_readfirstlane(
      (int)((((uint32_t)(gaddr >> 32)) & 0x01FFFFFFu) | (2u << 30)));
  // 2-operand form: VADDR2/VADDR3 = NULL (tensor up to 2D).
  asm volatile("tensor_load_to_lds %0, %1" :: "s"(g0), "s"(g1) : "memory");
}

__global__ __launch_bounds__(256) void yolo_main(
    const float* __restrict__ bb, const float* __restrict__ gt,
    const int* __restrict__ gsp, const int* __restrict__ imp,
    float* __restrict__ partials) {
  extern __shared__ char smem_raw[];
  float* sm = (float*)smem_raw;
  // LDS layout (floats): bb buf0 @0, bb buf1 @3072, gt buf0 @6144, gt buf1 @9216
  const unsigned tid = threadIdx.x;
  const int my_wave = __builtin_amdgcn_readfirstlane((int)(tid >> 5)); // SGPR
  const bool wave0 = (my_wave == 0);

  const float gs  = (float)gsp[0];
  const float im  = (float)imp[0];
  const float im1 = im - 1.0f;

  const uint32_t tile0  = blockIdx.x * TILES_PER_BLOCK;
  const uint64_t bbBase = (uint64_t)bb;
  const uint64_t gtBase = (uint64_t)gt;

  // D# group1: workgroup_mask=0, data_size=2 (4B), tensor_dim0=tile_dim0=3072,
  // tile_dim1=0 (1-D), no padding/iteration, strides 0.
  u32x8 g1;
  g1[0] = 0x00020000u;                        // data_size = 4 bytes
  g1[1] = (uint32_t)TILE_FLOATS << 16;        // tensor_dim0[15:0]
  g1[2] = 0u;
  g1[3] = (uint32_t)TILE_FLOATS << 16;        // tile_dim0
  g1[4] = 0u; g1[5] = 0u; g1[6] = 0u; g1[7] = 0u;

  if (wave0) {  // scalar branch: only wave 0 issues TDM (EXEC is ignored by TDM)
    tdm_load_1d(bbBase + (uint64_t)tile0 * TILE_BYTES, 0u, g1);
    tdm_load_1d(gtBase + (uint64_t)tile0 * TILE_BYTES, 2u * TILE_BYTES, g1);
  }

  float aNoOther = 0.f, aNoSel = 0.f, aConf = 0.f, aCoord = 0.f;
  float aCls = 0.f, aIou = 0.f, aCnt = 0.f;

  for (int k = 0; k < TILES_PER_BLOCK; ++k) {
    const uint32_t buf = (uint32_t)k & 1u;
    if (wave0) {
      if (k + 1 < TILES_PER_BLOCK) {
        const uint64_t off  = (uint64_t)(tile0 + (uint32_t)k + 1u) * TILE_BYTES;
        const uint32_t nbuf = ((uint32_t)k + 1u) & 1u;
        tdm_load_1d(bbBase + off, nbuf * TILE_BYTES, g1);
        tdm_load_1d(gtBase + off, 2u * TILE_BYTES + nbuf * TILE_BYTES, g1);
        // TDM completes in order: <=2 outstanding => tile k's 2 loads done.
        __builtin_amdgcn_s_wait_tensorcnt((short)2);
      } else {
        __builtin_amdgcn_s_wait_tensorcnt((short)0);
      }
    }
    __syncthreads();   // tile k resident in buf for all waves

    const float* bbp = sm + buf * TILE_FLOATS + tid * FLOATS_PER_CELL;
    const float* gtp = sm + 2u * TILE_FLOATS + buf * TILE_FLOATS +
                       tid * FLOATS_PER_CELL;
    float4 b0 = *(const float4*)(bbp);       // bb[0..3]
    float4 b1 = *(const float4*)(bbp + 4);   // bb[4..7]
    float4 b2 = *(const float4*)(bbp + 8);   // bb[8..11]
    float4 t0 = *(const float4*)(gtp);       // gt[0..3]
    float4 t1 = *(const float4*)(gtp + 4);   // gt[4..7]
    float4 t2 = *(const float4*)(gtp + 8);   // gt[8..11]

    const float c1 = b1.x, c2 = b2.y;
    const bool  sel2  = c1 < c2;
    const float p0    = sel2 ? b1.y : b0.x;
    const float p1    = sel2 ? b1.z : b0.y;
    const float p2    = sel2 ? b1.w : b0.z;
    const float p3    = sel2 ? b2.x : b0.w;
    const float selc  = sel2 ? c2 : c1;
    const float other = sel2 ? c1 : c2;

    const float objf   = (t2.y != 0.0f) ? 1.0f : 0.0f;
    const float noobjf = 1.0f - objf;

    aNoOther += other * other;
    aNoSel   += noobjf * selc * selc;
    const float dc = t1.x - selc;
    aConf    += objf * dc * dc;

    const float sg2 = sqrtf(fmaxf(t0.z, 0.0f) + 1e-08f);
    const float sg3 = sqrtf(fmaxf(t0.w, 0.0f) + 1e-08f);
    const float sp2 = sqrtf(fmaxf(p2,   0.0f) + 1e-08f);
    const float sp3 = sqrtf(fmaxf(p3,   0.0f) + 1e-08f);
    const float d0 = t0.x - p0, d1 = t0.y - p1, d2 = sg2 - sp2, d3 = sg3 - sp3;
    aCoord += objf * (d0 * d0 + d1 * d1 + d2 * d2 + d3 * d3);

    const float e0 = t2.z - b2.z, e1 = t2.w - b2.w;
    aCls += objf * (e0 * e0 + e1 * e1);

    // IoU vs truncated predicted box in image coordinates.
    const uint32_t g   = (tile0 + (uint32_t)k) * TILE_CELLS + tid;
    const uint32_t cim = g % CELLS_PER_IMG;
    const float gx = (float)(cim % S_DIM) * gs;
    const float gy = (float)(cim / S_DIM) * gs;
    const float px = truncf(gx + p0 * gs);
    const float py = truncf(gy + p1 * gs);
    const float pw = truncf(p2 * im);
    const float ph = truncf(p3 * im);
    const float pl = fmaxf(0.0f, px - pw * 0.5f);
    const float pu = fmaxf(0.0f, py - ph * 0.5f);
    const float pr = fminf(im1, px + pw * 0.5f);
    const float pd = fminf(im1, py + ph * 0.5f);
    const float pA = (pr - pl) * (pd - pu);
    const float gA = (t1.w - t1.y) * (t2.x - t1.z);      // (gr-gl)*(gd-gu)
    const float lx = fmaxf(pl, t1.y);
    const float rx = fminf(pr, t1.w);
    const float uy = fmaxf(pu, t1.z);
    const float dy = fminf(pd, t2.x);
    const float inter = (rx - lx) * (dy - uy);
    const float iou = ((rx < lx) || (dy < uy)) ? 0.0f
                      : inter / (pA + gA - inter);
    aIou += objf * iou;
    aCnt += objf;

    __syncthreads();   // reads of buf done before it is re-filled at k+2
  }

  // wave32 tree reduction
#pragma unroll
  for (int m = 16; m >= 1; m >>= 1) {
    aNoOther += __shfl_xor(aNoOther, m, 32);
    aNoSel   += __shfl_xor(aNoSel,   m, 32);
    aConf    += __shfl_xor(aConf,    m, 32);
    aCoord   += __shfl_xor(aCoord,   m, 32);
    aCls     += __shfl_xor(aCls,     m, 32);
    aIou     += __shfl_xor(aIou,     m, 32);
    aCnt     += __shfl_xor(aCnt,     m, 32);
  }
  __syncthreads();                 // LDS tile buffers are dead; reuse for waves
  if ((tid & 31u) == 0u) {
    const unsigned w = tid >> 5;
    sm[w * NACC + 0] = aNoOther; sm[w * NACC + 1] = aNoSel;
    sm[w * NACC + 2] = aConf;    sm[w * NACC + 3] = aCoord;
    sm[w * NACC + 4] = aCls;     sm[w * NACC + 5] = aIou;
    sm[w * NACC + 6] = aCnt;
  }
  __syncthreads();
  if (tid == 0) {
    float s[7] = {0.f, 0.f, 0.f, 0.f, 0.f, 0.f, 0.f};
    for (int w = 0; w < 8; ++w)
      for (int j = 0; j < 7; ++j) s[j] += sm[w * NACC + j];
    float* out = partials + (size_t)blockIdx.x * NACC;
    for (int j = 0; j < 7; ++j) out[j] = s[j];
    out[7] = 0.f;
  }
}

__global__ __launch_bounds__(256) void yolo_finalize(
    const float* __restrict__ partials, float* __restrict__ out) {
  __shared__ float red[8 * NACC];
  float s[7] = {0.f, 0.f, 0.f, 0.f, 0.f, 0.f, 0.f};
  for (unsigned b = threadIdx.x; b < MAIN_BLOCKS; b += 256u) {
    const float* p = partials + (size_t)b * NACC;
#pragma unroll
    for (int j = 0; j < 7; ++j) s[j] += p[j];
  }
#pragma unroll
  for (int m = 16; m >= 1; m >>= 1)
#pragma unroll
    for (int j = 0; j < 7; ++j) s[j] += __shfl_xor(s[j], m, 32);
  if ((threadIdx.x & 31u) == 0u) {
    const unsigned w = threadIdx.x >> 5;
#pragma unroll
    for (int j = 0; j < 7; ++j) red[w * NACC + j] = s[j];
  }
  __syncthreads();
  if (threadIdx.x == 0) {
    float t[7] = {0.f, 0.f, 0.f, 0.f, 0.f, 0.f, 0.f};
    for (int w = 0; w < 8; ++w)
      for (int j = 0; j < 7; ++j) t[j] += red[w * NACC + j];
    const float l_no   = 0.5f * (t[0] + t[1]);   // L_NOOBJ * (other + sel)
    const float l_conf = t[2];
    const float l_crd  = 5.0f * t[3];            // L_COORD
    const float l_cls  = t[4];
    out[0] = l_no + l_crd + l_conf + l_cls;      // loss
    out[1] = l_crd;                              // l_coord
    out[2] = l_no + l_conf;                      // loss_confidence
    out[3] = l_cls;                              // l_cls
    out[4] = t[5];                               // iou_sum
    out[5] = t[6];                               // object_num (as float value)
  }
}

extern "C" void kernel_launch(void* const* d_in, const int* in_sizes, int n_in,
                              void* d_out, int out_size, void* d_ws,
                              size_t ws_size, hipStream_t stream) {
  (void)in_sizes; (void)n_in; (void)out_size; (void)ws_size;
  const float* bb  = (const float*)d_in[0];
  const float* gt  = (const float*)d_in[1];
  const int*   gsp = (const int*)d_in[2];
  const int*   imp = (const int*)d_in[3];
  float* partials  = (float*)d_ws;               // MAIN_BLOCKS*NACC floats

  yolo_main<<<MAIN_BLOCKS, TILE_CELLS, 4 * TILE_BYTES, stream>>>(
      bb, gt, gsp, imp, partials);
  yolo_finalize<<<1, 256, 0, stream>>>(partials, (float*)d_out);
}